// DeeperGCN_19602230739362
// MI455X (gfx1250) — compile-verified
//
#include <hip/hip_runtime.h>
#include <math.h>

typedef __attribute__((ext_vector_type(2))) float v2f;
typedef __attribute__((ext_vector_type(8))) float v8f;

#define H    64
#define H2   128
#define NKP  17
#define NSEG 6
#define FDIM 256
#define EPSG 1e-7f
#define LNEPS 1e-5f

// ---- native CDNA5 no-return float atomics, device scope (cross-WGP coherent) ----
__device__ __forceinline__ void atom_max_f32(float* p, float v) {
  asm volatile("global_atomic_max_num_f32 %0, %1, off scope:SCOPE_DEV"
               : : "v"(p), "v"(v) : "memory");
}
__device__ __forceinline__ void atom_add_f32(float* p, float v) {
  asm volatile("global_atomic_add_f32 %0, %1, off scope:SCOPE_DEV"
               : : "v"(p), "v"(v) : "memory");
}

// ---------------- encoder: h = x[:, :2] @ enc_W + enc_b ----------------
__global__ void enc_kernel(const float* __restrict__ x, const float* __restrict__ W,
                           const float* __restrict__ b, float* __restrict__ h, int N) {
  int idx = blockIdx.x * blockDim.x + threadIdx.x;
  if (idx >= N * H) return;
  int n = idx >> 6, j = idx & 63;
  h[idx] = x[n * 3 + 0] * W[j] + x[n * 3 + 1] * W[H + j] + b[j];
}

// ------------- z = relu(LN(h, g, b)) ; one wave32 per node -------------
__global__ void prenorm_kernel(const float* __restrict__ h, const float* __restrict__ g,
                               const float* __restrict__ b, float* __restrict__ z, int N) {
  int node = (blockIdx.x * blockDim.x + threadIdx.x) >> 5;
  int lane = threadIdx.x & 31;
  if (node >= N) return;
  const float* hp = h + (size_t)node * H;
  float v0 = hp[lane], v1 = hp[lane + 32];
  float s = v0 + v1;
  for (int m = 16; m > 0; m >>= 1) s += __shfl_xor(s, m, 32);
  float mu = s * (1.f / 64.f);
  float d0 = v0 - mu, d1 = v1 - mu;
  float q = d0 * d0 + d1 * d1;
  for (int m = 16; m > 0; m >>= 1) q += __shfl_xor(q, m, 32);
  float inv = rsqrtf(q * (1.f / 64.f) + LNEPS);
  z[(size_t)node * H + lane]      = fmaxf(d0 * inv * g[lane]      + b[lane],      0.f);
  z[(size_t)node * H + lane + 32] = fmaxf(d1 * inv * g[lane + 32] + b[lane + 32], 0.f);
}

// ---------------- init segment buffers: m=-inf, den=num=0 ----------------
__global__ void init_seg_kernel(float* __restrict__ m, float* __restrict__ den,
                                float* __restrict__ num, int total) {
  int idx = blockIdx.x * blockDim.x + threadIdx.x;
  if (idx >= total) return;
  m[idx] = -__builtin_inff();
  den[idx] = 0.f;
  num[idx] = 0.f;
}

// ------ segment max over edges: one wave32 per edge, 2 feats per lane ------
__global__ void edge_max_kernel(const int* __restrict__ ei, const float* __restrict__ z,
                                const float* __restrict__ tptr, float* __restrict__ m, int E) {
  int tid = blockIdx.x * blockDim.x + threadIdx.x;
  int e = __builtin_amdgcn_readfirstlane(tid >> 5);   // wave-uniform edge id
  if (e >= E) return;
  int lane = threadIdx.x & 31;
  int src = ei[e], dst = ei[E + e];                   // scalar loads
  float t = tptr[0];
  float2 zz = *(const float2*)(z + (size_t)src * H + lane * 2);
  float* mp = m + (size_t)dst * H + lane * 2;
  atom_max_f32(mp,     t * (fmaxf(zz.x, 0.f) + EPSG));
  atom_max_f32(mp + 1, t * (fmaxf(zz.y, 0.f) + EPSG));
}

__global__ void fixmax_kernel(float* __restrict__ m, int total) {
  int idx = blockIdx.x * blockDim.x + threadIdx.x;
  if (idx >= total) return;
  float v = m[idx];
  if (!__builtin_isfinite(v)) m[idx] = 0.f;
}

// ------ segment softmax sums: one wave32 per edge, 2 feats per lane ------
__global__ void edge_sum_kernel(const int* __restrict__ ei, const float* __restrict__ z,
                                const float* __restrict__ tptr, const float* __restrict__ m,
                                float* __restrict__ den, float* __restrict__ num, int E) {
  int tid = blockIdx.x * blockDim.x + threadIdx.x;
  int e = __builtin_amdgcn_readfirstlane(tid >> 5);
  if (e >= E) return;
  int lane = threadIdx.x & 31;
  int src = ei[e], dst = ei[E + e];
  float t = tptr[0];
  float2 zz = *(const float2*)(z + (size_t)src * H + lane * 2);
  float2 mm = *(const float2*)(m + (size_t)dst * H + lane * 2);
  float msg0 = fmaxf(zz.x, 0.f) + EPSG;
  float msg1 = fmaxf(zz.y, 0.f) + EPSG;
  float a0 = __expf(t * msg0 - mm.x);
  float a1 = __expf(t * msg1 - mm.y);
  size_t base = (size_t)dst * H + lane * 2;
  atom_add_f32(den + base,     a0);
  atom_add_f32(den + base + 1, a1);
  atom_add_f32(num + base,     a0 * msg0);
  atom_add_f32(num + base + 1, a1 * msg1);
}

// ---- fused GENConv MLP: y=aggr+z ; relu(LN(y@W1+b1)) @ W2 + b2 ; residual ----
// block = 256 threads (8 waves), tile = 16 nodes.  fp32 WMMA 16x16x4.
__global__ __launch_bounds__(256)
void mlp_wmma_kernel(const float* __restrict__ num, const float* __restrict__ den,
                     const float* __restrict__ zin, float* __restrict__ h,
                     const float* __restrict__ W1, const float* __restrict__ b1,
                     const float* __restrict__ lg, const float* __restrict__ lb,
                     const float* __restrict__ W2, const float* __restrict__ b2,
                     int residual) {
  __shared__ float Ya[16][68];    // 16 nodes x 64  (pad 68 -> conflict-free col reads)
  __shared__ float Yb[16][132];   // 16 nodes x 128 (pad 132)
  int tile0 = blockIdx.x * 16;
  int t = threadIdx.x;
  int w = t >> 5, lane = t & 31;
  int hi = lane >> 4, lc = lane & 15;

  // stage 0: Y = num/den' + zin  into LDS
  for (int i = 0; i < 4; ++i) {
    int idx = i * 256 + t;
    int r = idx >> 6, c0 = idx & 63;
    size_t g = (size_t)(tile0 + r) * H + c0;
    float d = den[g];
    float aggr = num[g] / (d > 0.f ? d : 1.f);
    Ya[r][c0] = aggr + zin[g];
  }
  __syncthreads();

  // stage 1: GEMM1 (16x64)@(64x128): wave w -> cols [16w, 16w+16)
  {
    int col = w * 16 + lc;
    v8f acc = {0.f, 0.f, 0.f, 0.f, 0.f, 0.f, 0.f, 0.f};
    for (int k = 0; k < 16; ++k) {
      int kb = k * 4 + hi * 2;
      v2f av = {Ya[lc][kb], Ya[lc][kb + 1]};
      v2f bv = {W1[(size_t)kb * H2 + col], W1[(size_t)(kb + 1) * H2 + col]};
      acc = __builtin_amdgcn_wmma_f32_16x16x4_f32(false, av, false, bv,
                                                  (short)0, acc, false, false);
    }
    float bias = b1[col];
#pragma unroll
    for (int i = 0; i < 8; ++i)
      Yb[i + hi * 8][col] = acc[i] + bias;
  }
  __syncthreads();

  // stage 2: LN(128) + relu per row; each wave handles 2 rows
  for (int rr = 0; rr < 2; ++rr) {
    int r = w * 2 + rr;
    float v[4];
    float s = 0.f;
#pragma unroll
    for (int i = 0; i < 4; ++i) { v[i] = Yb[r][lane * 4 + i]; s += v[i]; }
    for (int mm = 16; mm > 0; mm >>= 1) s += __shfl_xor(s, mm, 32);
    float mu = s * (1.f / 128.f);
    float q = 0.f;
#pragma unroll
    for (int i = 0; i < 4; ++i) { v[i] -= mu; q += v[i] * v[i]; }
    for (int mm = 16; mm > 0; mm >>= 1) q += __shfl_xor(q, mm, 32);
    float inv = rsqrtf(q * (1.f / 128.f) + LNEPS);
#pragma unroll
    for (int i = 0; i < 4; ++i) {
      int cc = lane * 4 + i;
      Yb[r][cc] = fmaxf(v[i] * inv * lg[cc] + lb[cc], 0.f);
    }
  }
  __syncthreads();

  // stage 3: GEMM2 (16x128)@(128x64): waves 0..3
  if (w < 4) {
    int col = w * 16 + lc;
    v8f acc = {0.f, 0.f, 0.f, 0.f, 0.f, 0.f, 0.f, 0.f};
    for (int k = 0; k < 32; ++k) {
      int kb = k * 4 + hi * 2;
      v2f av = {Yb[lc][kb], Yb[lc][kb + 1]};
      v2f bv = {W2[(size_t)kb * H + col], W2[(size_t)(kb + 1) * H + col]};
      acc = __builtin_amdgcn_wmma_f32_16x16x4_f32(false, av, false, bv,
                                                  (short)0, acc, false, false);
    }
    float bias = b2[col];
#pragma unroll
    for (int i = 0; i < 8; ++i) {
      int r = i + hi * 8;
      size_t g = (size_t)(tile0 + r) * H + col;
      float val = acc[i] + bias;
      h[g] = residual ? (h[g] + val) : val;
    }
  }
}

// ---- tail: group-mean pool -> per-segment FC(64x256) + BN, WMMA ----
__global__ __launch_bounds__(256)
void fc_wmma_kernel(const float* __restrict__ hfin, const float* __restrict__ fcW,
                    const float* __restrict__ fcb, const float* __restrict__ bng,
                    const float* __restrict__ bnb, const float* __restrict__ bnm,
                    const float* __restrict__ bnv, float* __restrict__ f, int NP) {
  __shared__ float P[16][68];
  const int ga[NSEG] = {0, 0, 11, 0, 5, 11};
  const int gb[NSEG] = {17, 11, 17, 5, 11, 17};
  int tile0 = blockIdx.x * 16;
  int s = blockIdx.y;
  int half = blockIdx.z;           // cols 0-127 / 128-255
  int t = threadIdx.x, w = t >> 5, lane = t & 31;
  int hi = lane >> 4, lc = lane & 15;
  int a0 = ga[s], b0 = gb[s];
  float invc = 1.f / (float)(b0 - a0);

  for (int i = 0; i < 4; ++i) {
    int idx = i * 256 + t;
    int r = idx >> 6, cc = idx & 63;
    int p = tile0 + r;
    float sum = 0.f;
    for (int kp = a0; kp < b0; ++kp)
      sum += hfin[((size_t)p * NKP + kp) * H + cc];
    P[r][cc] = sum * invc;
  }
  __syncthreads();

  int col = half * 128 + w * 16 + lc;
  const float* Wp = fcW + (size_t)s * H * FDIM;
  v8f acc = {0.f, 0.f, 0.f, 0.f, 0.f, 0.f, 0.f, 0.f};
  for (int k = 0; k < 16; ++k) {
    int kb = k * 4 + hi * 2;
    v2f av = {P[lc][kb], P[lc][kb + 1]};
    v2f bv = {Wp[(size_t)kb * FDIM + col], Wp[(size_t)(kb + 1) * FDIM + col]};
    acc = __builtin_amdgcn_wmma_f32_16x16x4_f32(false, av, false, bv,
                                                (short)0, acc, false, false);
  }
  int sc = s * FDIM + col;
  float bias = fcb[sc];
  float scale = rsqrtf(bnv[sc] + 1e-5f) * bng[sc];
#pragma unroll
  for (int i = 0; i < 8; ++i) {
    int p = tile0 + i + hi * 8;
    float val = acc[i] + bias;
    val = (val - bnm[sc]) * scale + bnb[sc];
    f[(size_t)p * (NSEG * FDIM) + sc] = val;
  }
}

// ---------------- row L2 normalize (1536 per patient) ----------------
__global__ __launch_bounds__(256)
void l2norm_kernel(const float* __restrict__ f, float* __restrict__ out) {
  __shared__ float red[256];
  int p = blockIdx.x, t = threadIdx.x;
  const float* fp = f + (size_t)p * 1536;
  float s = 0.f;
#pragma unroll
  for (int i = 0; i < 6; ++i) { float v = fp[t + i * 256]; s += v * v; }
  red[t] = s;
  __syncthreads();
  for (int st = 128; st > 0; st >>= 1) {
    if (t < st) red[t] += red[t + st];
    __syncthreads();
  }
  float inv = 1.f / fmaxf(sqrtf(red[0]), 1e-12f);
#pragma unroll
  for (int i = 0; i < 6; ++i) {
    int idx = t + i * 256;
    out[(size_t)p * 1536 + idx] = fp[idx] * inv;
  }
}

extern "C" void kernel_launch(void* const* d_in, const int* in_sizes, int n_in,
                              void* d_out, int out_size, void* d_ws, size_t ws_size,
                              hipStream_t stream) {
  const float* x    = (const float*)d_in[0];
  const int*   ei   = (const int*)d_in[1];
  const float* encW = (const float*)d_in[2];
  const float* encb = (const float*)d_in[3];
  const float* tarr = (const float*)d_in[4];
  const float* W1   = (const float*)d_in[5];
  const float* b1   = (const float*)d_in[6];
  const float* lg   = (const float*)d_in[7];
  const float* lb   = (const float*)d_in[8];
  const float* W2   = (const float*)d_in[9];
  const float* b2   = (const float*)d_in[10];
  const float* ng   = (const float*)d_in[11];
  const float* nb   = (const float*)d_in[12];
  const float* fcW  = (const float*)d_in[13];
  const float* fcb  = (const float*)d_in[14];
  const float* bng  = (const float*)d_in[15];
  const float* bnb  = (const float*)d_in[16];
  const float* bnm  = (const float*)d_in[17];
  const float* bnv  = (const float*)d_in[18];

  int N = in_sizes[0] / 3;
  int E = in_sizes[1] / 2;
  int L = in_sizes[4];
  int NP = N / NKP;
  int NH = N * H;

  float* ws = (float*)d_ws;
  size_t off = 0;
  float* h    = ws + off; off += (size_t)NH;
  float* z    = ws + off; off += (size_t)NH;
  float* mmax = ws + off; off += (size_t)NH;
  float* dden = ws + off; off += (size_t)NH;
  float* nnum = ws + off; off += (size_t)NH;
  float* fbuf = ws + off;

  dim3 b256(256);
  enc_kernel<<<(NH + 255) / 256, b256, 0, stream>>>(x, encW, encb, h, N);

  // one wave32 per edge
  int egrid = (int)(((long long)E * 32 + 255) / 256);

  for (int l = 0; l < L; ++l) {
    const float* zin;
    if (l == 0) {
      zin = h;
    } else {
      prenorm_kernel<<<(N + 7) / 8, b256, 0, stream>>>(
          h, ng + (size_t)l * H, nb + (size_t)l * H, z, N);
      zin = z;
    }
    init_seg_kernel<<<(NH + 255) / 256, b256, 0, stream>>>(mmax, dden, nnum, NH);
    edge_max_kernel<<<egrid, b256, 0, stream>>>(ei, zin, tarr + l, mmax, E);
    fixmax_kernel<<<(NH + 255) / 256, b256, 0, stream>>>(mmax, NH);
    edge_sum_kernel<<<egrid, b256, 0, stream>>>(ei, zin, tarr + l, mmax, dden, nnum, E);
    mlp_wmma_kernel<<<N / 16, b256, 0, stream>>>(
        nnum, dden, zin, h,
        W1 + (size_t)l * H * H2, b1 + (size_t)l * H2,
        lg + (size_t)l * H2,     lb + (size_t)l * H2,
        W2 + (size_t)l * H2 * H, b2 + (size_t)l * H,
        l == 0 ? 0 : 1);
  }

  // final relu(LN) with norm_g[0]/norm_b[0]
  prenorm_kernel<<<(N + 7) / 8, b256, 0, stream>>>(h, ng, nb, z, N);
  fc_wmma_kernel<<<dim3(NP / 16, NSEG, 2), b256, 0, stream>>>(
      z, fcW, fcb, bng, bnb, bnm, bnv, fbuf, NP);
  l2norm_kernel<<<NP, b256, 0, stream>>>(fbuf, (float*)d_out);
}